// GNN_4432406249964
// MI455X (gfx1250) — compile-verified
//
#include <hip/hip_runtime.h>

#define NN 50000
#define DD 96
#define TILE 16
#define WSTRIDE 98   // padded LDS row stride (dwords) for transposed W: 8B-aligned, conflict-free

typedef __attribute__((ext_vector_type(2))) float v2f;
typedef __attribute__((ext_vector_type(8))) float v8f;

// ---------------- utility ----------------
__global__ void gcn_zero_kernel(float* __restrict__ p, int n) {
    int i = blockIdx.x * blockDim.x + threadIdx.x;
    if (i < n) p[i] = 0.0f;
}

// deg[d] += 1 for each edge (self-loop +1 folded into dinv kernel)
__global__ void gcn_deg_kernel(const int* __restrict__ dst, float* __restrict__ deg, int E) {
    int e = blockIdx.x * blockDim.x + threadIdx.x;
    if (e < E) atomicAdd(&deg[dst[e]], 1.0f);
}

// deg -> deg^{-1/2} in place (self-loop guarantees deg >= 1)
__global__ void gcn_dinv_kernel(float* __restrict__ deg, int n) {
    int i = blockIdx.x * blockDim.x + threadIdx.x;
    if (i < n) deg[i] = rsqrtf(deg[i] + 1.0f);
}

// ---------------- WMMA GEMM: H[n x 96] = X[n x 96] * W[96 x 96] ----------------
// Block = 256 threads = 8 waves. Each wave owns a 16-row strip and all 6 column
// tiles (16x96). W staged TRANSPOSED in LDS so each B fragment {W[k][c],W[k+1][c]}
// is one aligned ds_load_b64. V_WMMA_F32_16X16X4_F32, K loop of 24 steps.
__global__ __launch_bounds__(256) void gcn_gemm_wmma_kernel(
        const float* __restrict__ X, const float* __restrict__ W,
        float* __restrict__ H, int nrows) {
    __shared__ float sWt[DD * WSTRIDE];
    for (int i = threadIdx.x; i < DD * DD; i += 256) {
        const int k = i / DD;           // row of W
        const int c = i - k * DD;       // col of W
        sWt[c * WSTRIDE + k] = W[i];    // transposed store
    }
    __syncthreads();

    const int wave = threadIdx.x >> 5;
    const int lane = threadIdx.x & 31;
    const int r0 = (blockIdx.x * 8 + wave) * TILE;
    if (r0 >= nrows) return;            // whole-wave uniform exit; EXEC stays all-1s

    const int hi   = lane >> 4;         // 0: lanes 0-15, 1: lanes 16-31
    const int koff = hi * 2;            // second half-wave holds K+2, K+3
    const int mrow = r0 + (lane & 15);  // A: M = lane%16
    const int ncol = lane & 15;         // B/C/D: N = lane%16

    v8f acc[6];
#pragma unroll
    for (int t = 0; t < 6; ++t) acc[t] = (v8f){0.f,0.f,0.f,0.f,0.f,0.f,0.f,0.f};

    const float* __restrict__ xrow = X + (long)mrow * DD + koff;
    // per-tile transposed-W column pointers (column c of W = contiguous K run)
    const float* bcol[6];
#pragma unroll
    for (int t = 0; t < 6; ++t) bcol[t] = sWt + (t * TILE + ncol) * WSTRIDE + koff;

#pragma unroll 4
    for (int k = 0; k < DD; k += 4) {
        v2f a = *(const v2f*)(xrow + k);        // global_load_b64: K=k+koff, k+koff+1
#pragma unroll
        for (int t = 0; t < 6; ++t) {
            v2f b = *(const v2f*)(bcol[t] + k); // ds_load_b64, 8B aligned
            acc[t] = __builtin_amdgcn_wmma_f32_16x16x4_f32(
                false, a, false, b, (short)0, acc[t], false, false);
        }
    }

#pragma unroll
    for (int t = 0; t < 6; ++t) {
        const int n0 = t * TILE;
#pragma unroll
        for (int j = 0; j < 8; ++j) {
            const int row = r0 + j + hi * 8;   // D vgpr j: M=j (lo lanes) / j+8 (hi lanes)
            H[(long)row * DD + n0 + ncol] = acc[t][j];
        }
    }
}

// ---------------- edge aggregation: agg[d] += dinv[s]*dinv[d] * h[s] ----------------
// One wave32 per edge; each lane handles 3 of the 96 columns.
__global__ __launch_bounds__(256) void gcn_aggregate_kernel(
        const int* __restrict__ src, const int* __restrict__ dst,
        const float* __restrict__ dinv, const float* __restrict__ h,
        float* __restrict__ agg, int E) {
    const int e = blockIdx.x * 8 + (threadIdx.x >> 5);
    const int lane = threadIdx.x & 31;
    if (e >= E) return;
    const int s = src[e];
    const int d = dst[e];
    const float norm = dinv[s] * dinv[d];
    const float* __restrict__ hs = h + (long)s * DD;
    float* __restrict__ ad = agg + (long)d * DD;
#pragma unroll
    for (int c = 0; c < 3; ++c) {
        const int col = lane + c * 32;
        atomicAdd(&ad[col], norm * hs[col]);
    }
}

// ---------------- finalize: out = relu(agg + dinv^2 * h_self + bias) ----------------
// (self-loop message folded in here; out may alias h_self / agg elementwise)
__global__ void gcn_finalize_kernel(const float* __restrict__ agg,
                                    const float* h_self,
                                    const float* __restrict__ dinv,
                                    const float* __restrict__ bias,
                                    float* out, int n) {
    int i = blockIdx.x * blockDim.x + threadIdx.x;
    if (i < n * DD) {
        const int row = i / DD;
        const int col = i - row * DD;
        const float di = dinv[row];
        const float v = agg[i] + di * di * h_self[i] + bias[col];
        out[i] = fmaxf(v, 0.0f);
    }
}

// ---------------- driver ----------------
extern "C" void kernel_launch(void* const* d_in, const int* in_sizes, int n_in,
                              void* d_out, int out_size, void* d_ws, size_t ws_size,
                              hipStream_t stream) {
    const float* x  = (const float*)d_in[0];
    const int*   ei = (const int*)d_in[1];   // [2, E] flat
    const float* W1 = (const float*)d_in[2];
    const float* b1 = (const float*)d_in[3];
    const float* W2 = (const float*)d_in[4];
    const float* b2 = (const float*)d_in[5];

    const int n = in_sizes[0] / DD;          // 50000
    const int E = in_sizes[1] / 2;           // 800000
    const int* src = ei;
    const int* dst = ei + E;

    float* dinv = (float*)d_ws;              // n floats
    float* h    = dinv + n;                  // n*DD floats
    float* agg  = h + (size_t)n * DD;        // n*DD floats
    float* out  = (float*)d_out;

    const int nd = n * DD;
    const dim3 blk(256);
    const int gridN    = (n + 255) / 256;
    const int gridND   = (nd + 255) / 256;
    const int gridE    = (E + 255) / 256;
    const int gridEW   = (E + 7) / 8;        // 8 waves/block, 1 edge/wave
    const int gridGEMM = (n + 127) / 128;    // 8 row-tiles of 16 per block

    // degree -> dinv
    gcn_zero_kernel<<<gridN, blk, 0, stream>>>(dinv, n);
    gcn_deg_kernel<<<gridE, blk, 0, stream>>>(dst, dinv, E);
    gcn_dinv_kernel<<<gridN, blk, 0, stream>>>(dinv, n);

    // ---- layer 1 ----
    gcn_zero_kernel<<<gridND, blk, 0, stream>>>(agg, nd);
    gcn_gemm_wmma_kernel<<<gridGEMM, blk, 0, stream>>>(x, W1, h, n);
    gcn_aggregate_kernel<<<gridEW, blk, 0, stream>>>(src, dst, dinv, h, agg, E);
    gcn_finalize_kernel<<<gridND, blk, 0, stream>>>(agg, h, dinv, b1, h, n); // y1 -> h

    // ---- layer 2 ----
    gcn_zero_kernel<<<gridND, blk, 0, stream>>>(agg, nd);
    gcn_gemm_wmma_kernel<<<gridGEMM, blk, 0, stream>>>(h, W2, out, n);
    gcn_aggregate_kernel<<<gridEW, blk, 0, stream>>>(src, dst, dinv, out, agg, E);
    gcn_finalize_kernel<<<gridND, blk, 0, stream>>>(agg, out, dinv, b2, out, n);
}